// GraphConvolution_78030965833910
// MI455X (gfx1250) — compile-verified
//
#include <hip/hip_runtime.h>
#include <hip/hip_bf16.h>

// GraphConvolution for MI455X (gfx1250, wave32).
//   Phase 1: support = x @ W + bias     via V_WMMA_F32_16X16X4_F32
//   Phase 2: out = 0
//   Phase 3: out[row] += val * support[col]   (atomic fp32 scatter, 1 wave/edge)

#define D_FEAT 128

typedef __attribute__((ext_vector_type(2))) float v2f;
typedef __attribute__((ext_vector_type(8))) float v8f;

// ---------------------------------------------------------------------------
// GEMM: one wave computes one 16x16 tile of support = x * W (+bias).
// A (16x4 f32) layout: lane L -> M = L%16;  VGPR0 = K=(L/16)*2, VGPR1 = K+1
// B (4x16 f32) layout: lane L -> N = L%16;  VGPR0 = K=(L/16)*2, VGPR1 = K+1
// C/D (16x16 f32):     lane L -> N = L%16;  VGPR j -> M = j + 8*(L/16)
// ---------------------------------------------------------------------------
__global__ __launch_bounds__(256) void gcn_gemm_wmma_f32(
    const float* __restrict__ x,       // [n_rows, 128]
    const float* __restrict__ w,       // [128, 128]
    const float* __restrict__ bias,    // [128]
    float* __restrict__ support,       // [n_rows, 128]
    int n_rows) {
  const int wave      = (blockIdx.x * (blockDim.x >> 5)) + (threadIdx.x >> 5);
  const int col_tiles = D_FEAT / 16;                    // 8
  const int row_tile  = wave / col_tiles;
  const int col_tile  = wave - row_tile * col_tiles;
  if (row_tile * 16 >= n_rows) return;                  // wave-uniform: EXEC stays full

  const int lane = threadIdx.x & 31;
  const int m    = lane & 15;                           // M for A, N for B/C/D
  const int half = lane >> 4;
  const int row0 = row_tile * 16;
  const int col0 = col_tile * 16;

  // clamp for (unused here: 100000 % 16 == 0) ragged tails — keeps loads safe
  const int arow_idx = min(row0 + m, n_rows - 1);
  const float* __restrict__ arow = x + (size_t)arow_idx * D_FEAT;
  const float* __restrict__ bcol = w + col0 + m;        // stride D_FEAT down K

  v8f acc0 = {};
  v8f acc1 = {};
  #pragma unroll
  for (int k = 0; k < D_FEAT; k += 8) {
    const int ka = k + half * 2;                        // this lane-half's K base
    // two independent WMMA chains to hide D->C accumulation latency
    v2f a0 = *(const v2f*)(arow + ka);                  // K=ka, ka+1 (contiguous)
    v2f b0;
    b0.x = bcol[(size_t)(ka    ) * D_FEAT];
    b0.y = bcol[(size_t)(ka + 1) * D_FEAT];
    acc0 = __builtin_amdgcn_wmma_f32_16x16x4_f32(false, a0, false, b0,
                                                 (short)0, acc0, false, false);
    v2f a1 = *(const v2f*)(arow + ka + 4);
    v2f b1;
    b1.x = bcol[(size_t)(ka + 4) * D_FEAT];
    b1.y = bcol[(size_t)(ka + 5) * D_FEAT];
    acc1 = __builtin_amdgcn_wmma_f32_16x16x4_f32(false, a1, false, b1,
                                                 (short)0, acc1, false, false);
  }

  const float bv = bias[col0 + m];
  #pragma unroll
  for (int j = 0; j < 8; ++j) {
    const int r = row0 + j + 8 * half;
    if (r < n_rows)
      support[(size_t)r * D_FEAT + col0 + m] = acc0[j] + acc1[j] + bv;
  }
}

// ---------------------------------------------------------------------------
// Zero the output (segment_sum identity); float4 stores.
// ---------------------------------------------------------------------------
__global__ __launch_bounds__(256) void gcn_zero_f4(float4* __restrict__ p, size_t n4) {
  size_t i = (size_t)blockIdx.x * blockDim.x + threadIdx.x;
  const size_t stride = (size_t)gridDim.x * blockDim.x;
  const float4 z = {0.f, 0.f, 0.f, 0.f};
  for (; i < n4; i += stride) p[i] = z;
}

// ---------------------------------------------------------------------------
// Scatter: one wave per edge. Lane l handles channels [4l, 4l+3].
// out[row] += val * support[col]  via global_atomic_add_f32.
// ---------------------------------------------------------------------------
__global__ __launch_bounds__(256) void gcn_edge_scatter(
    const float* __restrict__ support,
    const float* __restrict__ edge_val,
    const int*   __restrict__ edge_row,
    const int*   __restrict__ edge_col,
    float* __restrict__ out,
    int n_edges) {
  const int wave = (blockIdx.x * (blockDim.x >> 5)) + (threadIdx.x >> 5);
  if (wave >= n_edges) return;                          // wave-uniform
  const int lane = threadIdx.x & 31;

  const int   src = edge_col[wave];                     // wave-uniform -> scalar loads
  const int   dst = edge_row[wave];
  const float v   = edge_val[wave];

  const float4 s = ((const float4*)(support + (size_t)src * D_FEAT))[lane];
  float* __restrict__ o = out + (size_t)dst * D_FEAT + lane * 4;
  atomicAdd(o + 0, v * s.x);
  atomicAdd(o + 1, v * s.y);
  atomicAdd(o + 2, v * s.z);
  atomicAdd(o + 3, v * s.w);
}

// ---------------------------------------------------------------------------
extern "C" void kernel_launch(void* const* d_in, const int* in_sizes, int n_in,
                              void* d_out, int out_size, void* d_ws, size_t ws_size,
                              hipStream_t stream) {
  const float* x        = (const float*)d_in[0];
  const float* edge_val = (const float*)d_in[1];
  const float* weight   = (const float*)d_in[2];
  const float* bias     = (const float*)d_in[3];
  const int*   edge_row = (const int*)d_in[4];
  const int*   edge_col = (const int*)d_in[5];
  float*       out      = (float*)d_out;
  float*       support  = (float*)d_ws;   // needs n_nodes*128*4 = 51.2 MB

  const int n_nodes = in_sizes[0] / D_FEAT;   // 100000
  const int n_edges = in_sizes[1];            // 3200000

  // Phase 2 first (independent of GEMM): clear output
  {
    const size_t n4 = (size_t)n_nodes * D_FEAT / 4;
    const int blocks = (int)((n4 + 255) / 256);
    gcn_zero_f4<<<blocks, 256, 0, stream>>>((float4*)out, n4);
  }

  // Phase 1: support = x @ W + bias  (WMMA fp32)
  {
    const int row_tiles   = (n_nodes + 15) / 16;        // 6250
    const int total_waves = row_tiles * (D_FEAT / 16);  // 50000
    const int blocks      = (total_waves + 7) / 8;      // 8 waves / 256-thread block
    gcn_gemm_wmma_f32<<<blocks, 256, 0, stream>>>(x, weight, bias, support, n_nodes);
  }

  // Phase 3: edge scatter-add
  {
    const int blocks = (n_edges + 7) / 8;               // 8 edges (waves) / block
    gcn_edge_scatter<<<blocks, 256, 0, stream>>>(support, edge_val, edge_row,
                                                 edge_col, out, n_edges);
  }
}